// NN_17_42820823941691
// MI455X (gfx1250) — compile-verified
//
#include <hip/hip_runtime.h>

// FSRS-style recurrent scan: T=64 steps, B=131072 columns, 2-float state.
// One thread per column, state in VGPRs, fp32 VALU with native exp2/log2/rcp.
// WMMA is inapplicable (hidden dim == 1, no matmul); the kernel is a balanced
// HBM-stream + transcendental-VALU workload (~169 MB traffic, ~7 us floor).
// Outputs are write-once -> non-temporal stores; inputs are prefetched 4
// steps ahead with gfx1250 global_prefetch_b8.

#define T_STEPS 64

typedef float v2f __attribute__((ext_vector_type(2)));

__device__ __forceinline__ float fexp(float x) {           // e^x
    return __builtin_amdgcn_exp2f(x * 1.4426950408889634f);
}
__device__ __forceinline__ float flog(float x) {           // ln x
    return __builtin_amdgcn_logf(x) * 0.6931471805599453f;
}
__device__ __forceinline__ float frcp(float x) {
    return __builtin_amdgcn_rcpf(x);
}
// Numerically stable softplus: max(x,0) + log1p(exp(-|x|))
__device__ __forceinline__ float softplus(float x) {
    float e = __builtin_amdgcn_exp2f(-fabsf(x) * 1.4426950408889634f);
    return fmaxf(x, 0.0f) + flog(1.0f + e);
}
// tanh(y) for y >= 0, overflow-safe: 1 - 2/(exp(2y)+1)
__device__ __forceinline__ float tanh_nn(float y) {
    float e = __builtin_amdgcn_exp2f(y * 2.8853900817779268f); // exp(2y)
    return 1.0f - 2.0f * frcp(e + 1.0f);
}
__device__ __forceinline__ float mish(float x) {
    return x * tanh_nn(softplus(x));
}
__device__ __forceinline__ float sigmoid(float x) {
    return frcp(1.0f + fexp(-x));
}
__device__ __forceinline__ float clipf(float x, float lo, float hi) {
    return fminf(fmaxf(x, lo), hi);
}
__device__ __forceinline__ void store_nt2(float* p, float x, float y) {
    v2f v; v.x = x; v.y = y;
    __builtin_nontemporal_store(v, (v2f*)p);
}

__global__ __launch_bounds__(256) void fsrs_scan_kernel(
    const float* __restrict__ inp,   // (T, B, 3): dt, rating, lapses
    const float* __restrict__ S0p, const float* __restrict__ D0p,
    const float* __restrict__ swp,
    const float* __restrict__ rw_w1, const float* __restrict__ rw_b1,
    const float* __restrict__ rw_w2, const float* __restrict__ rw_b2,
    const float* __restrict__ nd_w1, const float* __restrict__ nd_b1,
    const float* __restrict__ nd_w2, const float* __restrict__ nd_b2,
    const float* __restrict__ pl_w1, const float* __restrict__ pl_b1,
    const float* __restrict__ pl_w2, const float* __restrict__ pl_b2,
    const float* __restrict__ sn_w1, const float* __restrict__ sn_b1,
    const float* __restrict__ sn_w2, const float* __restrict__ sn_b2,
    const float* __restrict__ bs_w1, const float* __restrict__ bs_b1,
    const float* __restrict__ bs_w2, const float* __restrict__ bs_b2,
    float* __restrict__ out, int B)
{
    const int b = blockIdx.x * blockDim.x + threadIdx.x;
    if (b >= B) return;

    // ---- uniform parameters: SGPR-resident scalar loads, hoisted ----
    const float s00 = S0p[0], s01 = S0p[1], s02 = S0p[2], s03 = S0p[3];
    const float d00 = D0p[0], d01 = D0p[1], d02 = D0p[2], d03 = D0p[3];
    const float sw0 = swp[0], sw1 = swp[1], sw2 = swp[2];
    const float esw0 = fexp(sw0);

    const float rA0 = rw_w1[0], rA1 = rw_w1[1], rA2 = rw_w1[2];
    const float rB1 = rw_b1[0], rW2 = rw_w2[0], rB2 = rw_b2[0];
    const float nA0 = nd_w1[0], nA1 = nd_w1[1], nA2 = nd_w1[2];
    const float nB1 = nd_b1[0], nW2 = nd_w2[0], nB2 = nd_b2[0];
    const float pA0 = pl_w1[0], pA1 = pl_w1[1];
    const float pB1 = pl_b1[0], pW2 = pl_w2[0], pB2 = pl_b2[0];
    const float tA0 = sn_w1[0], tA1 = sn_w1[1], tA2 = sn_w1[2];
    const float tB1 = sn_b1[0], tW2 = sn_w2[0], tB2 = sn_b2[0];
    const float bA0 = bs_w1[0], bA1 = bs_w1[1];
    const float bB1 = bs_b1[0], bW2 = bs_w2[0], bB2 = bs_b2[0];

    const float LOG09   = -0.10536051565782628f;   // ln(0.9)
    const float LOG2_09 = -0.15200309344504995f;   // log2(0.9)
    const float S_MIN = 0.01f, S_MAX = 36500.0f;

    const size_t lane_off = (size_t)b * 3;
    const size_t row3 = (size_t)B * 3;             // floats per timestep (inputs)

    // ---- t = 0: init branch (state is exactly all-ones only here) ----
    float s, d;
    {
        const float* xp = inp + lane_off;
        float rating = xp[1];
        int idx = (int)rating - 1;                 // rating in [1,4] always valid
        s = (idx == 0) ? s00 : (idx == 1) ? s01 : (idx == 2) ? s02 : s03;
        d = (idx == 0) ? d00 : (idx == 1) ? d01 : (idx == 2) ? d02 : d03;
        s = clipf(s, S_MIN, S_MAX);
        d = clipf(d, 0.0001f, 1.0f);
        store_nt2(out + (size_t)b * 2, s, d);
    }

    // ---- t = 1 .. 63: update branch ----
    for (int t = 1; t < T_STEPS; ++t) {
        const float* xp = inp + (size_t)t * row3 + lane_off;
        float dt     = xp[0];
        float rating = xp[1];
        float lapses = xp[2];

        // gfx1250 prefetch (global_prefetch_b8): keep the stream deep in flight
        if (t + 4 < T_STEPS) __builtin_prefetch(xp + 4 * row3, 0, 1);

        // rt = clip(0.9^(dt/s), 1e-4, 0.9999)
        float rt = clipf(__builtin_amdgcn_exp2f(dt * frcp(s) * LOG2_09),
                         0.0001f, 0.9999f);

        // rw = clip(exp(-softplus(mlp_rw([d, s, rt]))), 1e-4, 0.9999)
        float hr = fmaf(rA0, d, fmaf(rA1, s, fmaf(rA2, rt, rB1)));
        float yr = fmaf(mish(hr), rW2, rB2);
        float rw = clipf(fexp(-softplus(yr)), 0.0001f, 0.9999f);

        // sr = clip(ln(0.9)/ln(rw) * dt, S_MIN, S_MAX)
        float sr = clipf(LOG09 * frcp(flog(rw)) * dt, S_MIN, S_MAX);

        // new_d = sigmoid(mlp_nd([d, rw, rating]))
        float hn = fmaf(nA0, d, fmaf(nA1, rw, fmaf(nA2, rating, nB1)));
        float ndv = sigmoid(fmaf(mish(hn), nW2, nB2));

        // pls = clip(softplus(mlp_pls([rw, lapses])), S_MIN, S_MAX)
        float hp = fmaf(pA0, rw, fmaf(pA1, lapses, pB1));
        float pls = clipf(softplus(fmaf(mish(hp), pW2, pB2)), S_MIN, S_MAX);

        // sinc_t = 1 + e^sw0 * (5(1-nd)+1) * sr^(-sw1) * e^(-rw*sw2)
        float powv = __builtin_amdgcn_exp2f(-sw1 * __builtin_amdgcn_logf(sr));
        float sinc_t = fmaf(esw0 * fmaf(-5.0f, ndv, 6.0f) * powv,
                            fexp(-rw * sw2), 1.0f);

        // sinc_nn = 1 + softplus(mlp_snn([nd, sr, rw]))
        float ht = fmaf(tA0, ndv, fmaf(tA1, sr, fmaf(tA2, rw, tB1)));
        float sinc_nn = 1.0f + softplus(fmaf(mish(ht), tW2, tB2));

        // best = clip(1 + softplus(mlp_bs([sinc_t, sinc_nn])), 1, 100)
        float hb = fmaf(bA0, sinc_t, fmaf(bA1, sinc_nn, bB1));
        float best = clipf(1.0f + softplus(fmaf(mish(hb), bW2, bB2)),
                           1.0f, 100.0f);

        float new_s = (rating > 1.0f) ? sr * best : pls;
        s = clipf(new_s, S_MIN, S_MAX);
        d = clipf(ndv, 0.0001f, 1.0f);

        store_nt2(out + ((size_t)t * B + b) * 2, s, d);
    }

    // final_state, concatenated after outputs: offset T*B*2 + b*2
    store_nt2(out + (size_t)T_STEPS * B * 2 + (size_t)b * 2, s, d);
}

extern "C" void kernel_launch(void* const* d_in, const int* in_sizes, int n_in,
                              void* d_out, int out_size, void* d_ws, size_t ws_size,
                              hipStream_t stream) {
    (void)d_ws; (void)ws_size; (void)out_size;
    auto f = [&](int i) { return (const float*)d_in[i]; };

    const float* inp = f(0);
    const float* S0  = f(1);
    const float* D0  = f(2);
    const float* sw  = f(3);

    const float *rw1, *rb1, *rw2, *rb2;
    const float *nw1, *nb1, *nw2, *nb2;
    const float *pw1, *pb1, *pw2, *pb2;
    const float *sn1, *sb1, *sn2, *sb2;
    const float *bw1, *bb1, *bw2, *bb2;

    if (n_in >= 24) {
        // tuples delivered as separate pointers: (w1,b1,w2,b2) x 5
        rw1 = f(4);  rb1 = f(5);  rw2 = f(6);  rb2 = f(7);
        nw1 = f(8);  nb1 = f(9);  nw2 = f(10); nb2 = f(11);
        pw1 = f(12); pb1 = f(13); pw2 = f(14); pb2 = f(15);
        sn1 = f(16); sb1 = f(17); sn2 = f(18); sb2 = f(19);
        bw1 = f(20); bb1 = f(21); bw2 = f(22); bb2 = f(23);
    } else {
        // tuples flattened: rw(6)=w1[3],b1,w2,b2 ; pls/bs have din=2 -> 5 floats
        const float* p;
        p = f(4); rw1 = p; rb1 = p + 3; rw2 = p + 4; rb2 = p + 5;
        p = f(5); nw1 = p; nb1 = p + 3; nw2 = p + 4; nb2 = p + 5;
        p = f(6); pw1 = p; pb1 = p + 2; pw2 = p + 3; pb2 = p + 4;
        p = f(7); sn1 = p; sb1 = p + 3; sn2 = p + 4; sb2 = p + 5;
        p = f(8); bw1 = p; bb1 = p + 2; bw2 = p + 3; bb2 = p + 4;
    }

    const int B = in_sizes[0] / (T_STEPS * 3);   // 131072
    const int block = 256;
    const int grid = (B + block - 1) / block;

    fsrs_scan_kernel<<<grid, block, 0, stream>>>(
        inp, S0, D0, sw,
        rw1, rb1, rw2, rb2,
        nw1, nb1, nw2, nb2,
        pw1, pb1, pw2, pb2,
        sn1, sb1, sn2, sb2,
        bw1, bb1, bw2, bb2,
        (float*)d_out, B);
}